// _FallbackGCNConv_47364899340492
// MI455X (gfx1250) — compile-verified
//
#include <hip/hip_runtime.h>
#include <hip/hip_bf16.h>

typedef __attribute__((ext_vector_type(2))) float v2f;
typedef __attribute__((ext_vector_type(8))) float v8f;

// ---------------------------------------------------------------------------
// 1) Degree: deg[i] = 1 (self loop) + sum_{e: row==i} w[e];   dinv = rsqrt(deg)
// ---------------------------------------------------------------------------
__global__ void k_deg_init(float* __restrict__ deg, int n) {
    int i = blockIdx.x * blockDim.x + threadIdx.x;
    if (i < n) deg[i] = 1.0f;  // self-loop weight
}

__global__ void k_deg_scatter(const long long* __restrict__ ei,
                              const float* __restrict__ ew,
                              float* __restrict__ deg, int E) {
    int e = blockIdx.x * blockDim.x + threadIdx.x;
    if (e < E) atomicAdd(&deg[(int)ei[e]], ew[e]);   // ei[0][e] = row
}

__global__ void k_deg_rsqrt(float* __restrict__ deg, int n) {
    int i = blockIdx.x * blockDim.x + threadIdx.x;
    if (i < n) deg[i] = rsqrtf(deg[i]);              // deg >= 1 always
}

// ---------------------------------------------------------------------------
// 2a) Self-loop init of aggregate (written directly into d_out, in-place GEMM
//     input):  agg[i][d] = x[i][d] * dinv[i]^2
// ---------------------------------------------------------------------------
__global__ void k_agg_init(const float* __restrict__ x,
                           const float* __restrict__ dinv,
                           float* __restrict__ agg, int total) {
    int idx = blockIdx.x * blockDim.x + threadIdx.x;
    if (idx < total) {
        float s = dinv[idx >> 6];
        agg[idx] = x[idx] * s * s;
    }
}

// ---------------------------------------------------------------------------
// 2b) Edge aggregation: one wave32 per edge, float2 per lane (64 cols).
//     agg[row] += x[col] * (dinv[row] * w * dinv[col])   via atomic f32 adds
// ---------------------------------------------------------------------------
__global__ void k_edge_agg(const float* __restrict__ x,
                           const long long* __restrict__ ei,
                           const float* __restrict__ ew,
                           const float* __restrict__ dinv,
                           float* __restrict__ agg, int E) {
    int e    = blockIdx.x * 8 + (threadIdx.x >> 5);   // 8 waves / block
    int lane = threadIdx.x & 31;
    if (e >= E) return;
    long long r = ei[e];                 // row
    long long c = ei[(size_t)E + e];     // col
    float norm = dinv[r] * ew[e] * dinv[c];
    float2 xv = ((const float2*)(x + (size_t)c * 64))[lane];
    float* ap = agg + (size_t)r * 64 + lane * 2;
    atomicAdd(ap + 0, xv.x * norm);
    atomicAdd(ap + 1, xv.y * norm);
}

// ---------------------------------------------------------------------------
// 3) Projection: out = agg @ W^T + bias, f32 WMMA 16x16x4, in-place in d_out.
//    One wave computes a 16-row x 64-col stripe (4 column tiles, K=64 in 16
//    steps). W is staged in LDS with row pitch 68 floats (conflict-free b64).
// ---------------------------------------------------------------------------
#define WPITCH 68

__global__ void k_gemm_wmma(float* __restrict__ out,       // agg in, result out
                            const float* __restrict__ W,   // [64][64] row-major
                            const float* __restrict__ bias,
                            int nRows) {
    __shared__ float ldsW[64 * WPITCH];
    int tid = threadIdx.x;

    // cooperative load of W (64x64 f32 = 16KB) into padded LDS
    for (int idx = tid; idx < 64 * 64; idx += 256) {
        int n = idx >> 6, k = idx & 63;
        ldsW[n * WPITCH + k] = W[idx];
    }
    __syncthreads();

    int wave = tid >> 5;
    int lane = tid & 31;
    int tile = blockIdx.x * 8 + wave;
    int m0   = tile * 16;
    if (m0 >= nRows) return;

    int half = lane >> 4;     // 0: K pair {0,1}, 1: K pair {2,3}
    int l16  = lane & 15;

    // A fragment base: row m0+l16, element (4*kb + 2*half)
    const float* aBase = out + (size_t)(m0 + l16) * 64 + 2 * half;
    // B fragment bases: B[k][n] = W[n][k]; col tile t -> W row (16t + l16)
    const float* bB0 = ldsW + (0 * 16 + l16) * WPITCH + 2 * half;
    const float* bB1 = ldsW + (1 * 16 + l16) * WPITCH + 2 * half;
    const float* bB2 = ldsW + (2 * 16 + l16) * WPITCH + 2 * half;
    const float* bB3 = ldsW + (3 * 16 + l16) * WPITCH + 2 * half;

    v8f c0 = {}, c1 = {}, c2 = {}, c3 = {};

#pragma unroll
    for (int kb = 0; kb < 16; ++kb) {
        int ko = kb * 4;
        v2f a  = *(const v2f*)(aBase + ko);
        v2f b0 = *(const v2f*)(bB0 + ko);
        v2f b1 = *(const v2f*)(bB1 + ko);
        v2f b2 = *(const v2f*)(bB2 + ko);
        v2f b3 = *(const v2f*)(bB3 + ko);
        c0 = __builtin_amdgcn_wmma_f32_16x16x4_f32(false, a, false, b0, (short)0, c0, false, false);
        c1 = __builtin_amdgcn_wmma_f32_16x16x4_f32(false, a, false, b1, (short)0, c1, false, false);
        c2 = __builtin_amdgcn_wmma_f32_16x16x4_f32(false, a, false, b2, (short)0, c2, false, false);
        c3 = __builtin_amdgcn_wmma_f32_16x16x4_f32(false, a, false, b3, (short)0, c3, false, false);
    }

    float bv0 = bias[ 0 + l16];
    float bv1 = bias[16 + l16];
    float bv2 = bias[32 + l16];
    float bv3 = bias[48 + l16];

    // C/D layout: VGPR r, lanes 0-15 -> row m0+r, lanes 16-31 -> row m0+8+r
#pragma unroll
    for (int r = 0; r < 8; ++r) {
        float* o = out + (size_t)(m0 + r + 8 * half) * 64;
        o[ 0 + l16] = c0[r] + bv0;
        o[16 + l16] = c1[r] + bv1;
        o[32 + l16] = c2[r] + bv2;
        o[48 + l16] = c3[r] + bv3;
    }
}

// ---------------------------------------------------------------------------
// Launch: x(0) [N,64] f32 | edge_index(1) [2,E] i64 | edge_weight(2) [E] f32 |
//         W(3) [64,64] f32 | bias(4) [64] f32   ->  d_out [N,64] f32
// ---------------------------------------------------------------------------
extern "C" void kernel_launch(void* const* d_in, const int* in_sizes, int n_in,
                              void* d_out, int out_size, void* d_ws, size_t ws_size,
                              hipStream_t stream) {
    const float*     x    = (const float*)d_in[0];
    const long long* ei   = (const long long*)d_in[1];
    const float*     ew   = (const float*)d_in[2];
    const float*     W    = (const float*)d_in[3];
    const float*     bias = (const float*)d_in[4];
    float*           out  = (float*)d_out;

    const int N = in_sizes[0] / 64;     // 100000
    const int E = in_sizes[2];          // 1000000

    float* deg = (float*)d_ws;          // N floats; becomes dinv in-place

    // 1) degree -> dinv
    k_deg_init   <<<(N + 255) / 256, 256, 0, stream>>>(deg, N);
    k_deg_scatter<<<(E + 255) / 256, 256, 0, stream>>>(ei, ew, deg, E);
    k_deg_rsqrt  <<<(N + 255) / 256, 256, 0, stream>>>(deg, N);

    // 2) aggregate into d_out (self loops + edge scatter)
    k_agg_init<<<(N * 64 + 255) / 256, 256, 0, stream>>>(x, deg, out, N * 64);
    k_edge_agg<<<(E + 7) / 8, 256, 0, stream>>>(x, ei, ew, deg, out, E);

    // 3) in-place projection with f32 WMMA
    int tiles = (N + 15) / 16;
    k_gemm_wmma<<<(tiles + 7) / 8, 256, 0, stream>>>(out, W, bias, N);
}